// DiffJPEG_78323023610387
// MI455X (gfx1250) — compile-verified
//
#include <hip/hip_runtime.h>
#include <math.h>

typedef float v2f __attribute__((ext_vector_type(2)));
typedef float v8f __attribute__((ext_vector_type(8)));

__device__ __forceinline__ float poly_floor_f(float x) {
    float f = floorf(x);
    float d = x - 0.5f - f;
    return f + d * d * d;
}

__device__ __forceinline__ float diff_clip_f(float x) {
    const float mn = 1.0f, mx = 255.0f, sc = 0.02f;
    if (x > mx) x = -sc * (expf(mx - x) - 1.0f) + mx;
    if (x < mn) x =  sc * (expf(x - mn) - 1.0f) + mn;
    return x;
}

// CA_TAB[x*8+u] = alpha[x] * cos((2u+1)*x*pi/16), alpha[0] = 1/sqrt(2).
// Precomputed so the hot IDCT kernel does table loads instead of cosf chains.
__device__ const float CA_TAB[64] = {
    // x = 0
     0.7071067812f,  0.7071067812f,  0.7071067812f,  0.7071067812f,
     0.7071067812f,  0.7071067812f,  0.7071067812f,  0.7071067812f,
    // x = 1
     0.9807852804f,  0.8314696123f,  0.5555702330f,  0.1950903220f,
    -0.1950903220f, -0.5555702330f, -0.8314696123f, -0.9807852804f,
    // x = 2
     0.9238795325f,  0.3826834324f, -0.3826834324f, -0.9238795325f,
    -0.9238795325f, -0.3826834324f,  0.3826834324f,  0.9238795325f,
    // x = 3
     0.8314696123f, -0.1950903220f, -0.9807852804f, -0.5555702330f,
     0.5555702330f,  0.9807852804f,  0.1950903220f, -0.8314696123f,
    // x = 4
     0.7071067812f, -0.7071067812f, -0.7071067812f,  0.7071067812f,
     0.7071067812f, -0.7071067812f, -0.7071067812f,  0.7071067812f,
    // x = 5
     0.5555702330f, -0.9807852804f,  0.1950903220f,  0.8314696123f,
    -0.8314696123f, -0.1950903220f,  0.9807852804f, -0.5555702330f,
    // x = 6
     0.3826834324f, -0.9238795325f,  0.9238795325f, -0.3826834324f,
    -0.3826834324f,  0.9238795325f, -0.9238795325f,  0.3826834324f,
    // x = 7
     0.1950903220f, -0.5555702330f,  0.8314696123f, -0.9807852804f,
     0.9807852804f, -0.8314696123f,  0.5555702330f, -0.1950903220f
};

// ---------------------------------------------------------------------------
// Kernel 0: per-batch dequantization multiplier tables (64 entries each).
// mult[i*8+j] = poly_floor(diff_clip((qt[i][j]*scale(q) + 50)/100, 1, 255))
// ---------------------------------------------------------------------------
__global__ void mult_kernel(const float* __restrict__ q,
                            const float* __restrict__ qty,
                            const float* __restrict__ qtc,
                            float* __restrict__ multY,
                            float* __restrict__ multC) {
    int b = blockIdx.x;
    int t = threadIdx.x;
    float qq = q[b];
    float s = poly_floor_f((qq < 50.0f) ? (5000.0f / qq) : (200.0f - 2.0f * qq));
    if (t < 64) {
        float v = (qty[t] * s + 50.0f) * 0.01f;
        multY[b * 64 + t] = poly_floor_f(diff_clip_f(v));
    } else {
        int u = t - 64;
        float v = (qtc[u] * s + 50.0f) * 0.01f;
        multC[b * 64 + u] = poly_floor_f(diff_clip_f(v));
    }
}

// ---------------------------------------------------------------------------
// Kernel 1/2/3: dequantize + 8x8 IDCT via V_WMMA_F32_16X16X4_F32.
// Each wave handles 4 JPEG blocks packed as a 16x16 tile [[Z1,Z2],[Z3,Z4]].
// Stage 1: P = diag(Ca^T,Ca^T) x tile   (4 chained K=4 WMMAs)
// Stage 2: D = P x diag(Ca,Ca)          (4 chained K=4 WMMAs, A via LDS)
// Result (0.25*D + 128)/255 written unpatchified to a plane.
// ---------------------------------------------------------------------------
__global__ void __launch_bounds__(256)
idct8_wmma_kernel(const float* __restrict__ in,    // [B, Nblk, 8, 8]
                  const float* __restrict__ mult,  // [B, 64]
                  float* __restrict__ outp,
                  int Nblk, int Wpix,
                  long batchStride, int replicate, long chStride,
                  long totalTiles) {
    __shared__ float pbuf[8 * 272];  // per-wave 16x17 padded tile

    int lane = threadIdx.x & 31;
    int wv   = threadIdx.x >> 5;
    long tile = (long)blockIdx.x * 8 + wv;
    bool valid = tile < totalTiles;
    if (!valid) tile = totalTiles - 1;  // clamp: keep EXEC all-ones for WMMA

    int tpb = Nblk >> 2;                 // tiles per batch
    int b   = (int)(tile / tpb);
    int n0  = (int)(tile % tpb) * 4;     // first of 4 consecutive blocks
    const float* blkBase = in + ((long)b * Nblk + n0) * 64;
    const float* mb      = mult + (long)b * 64;

    int ln   = lane & 15;   // A-row / B-col / D-col index
    int half = lane >> 4;

    // Constant operand: serves as stage-1 A (diag(Ca^T) in A-layout) and
    // stage-2 B (diag(Ca) in B-layout) — identical per-lane values.
    v2f cmat[4];
#pragma unroll
    for (int k = 0; k < 4; k++) {
        int c0 = 4 * k + 2 * half;       // even, so c0 and c0+1 share a half
        float g = ((ln >> 3) == (c0 >> 3)) ? 1.0f : 0.0f;
        cmat[k].x = g * CA_TAB[(c0 & 7) * 8 + (ln & 7)];
        cmat[k].y = g * CA_TAB[((c0 + 1) & 7) * 8 + (ln & 7)];
    }

    // Stage 1: load dequantized tile rows as B operand, accumulate P.
    v8f acc = {0.f, 0.f, 0.f, 0.f, 0.f, 0.f, 0.f, 0.f};
#pragma unroll
    for (int k = 0; k < 4; k++) {
        int r0 = 4 * k + 2 * half;
        v2f bv;
        {
            int blk = ((r0 >> 3) << 1) + (ln >> 3);
            int i = r0 & 7, j = ln & 7;
            bv.x = blkBase[blk * 64 + i * 8 + j] * mb[i * 8 + j];
        }
        {
            int r1 = r0 + 1;
            int blk = ((r1 >> 3) << 1) + (ln >> 3);
            int i = r1 & 7, j = ln & 7;
            bv.y = blkBase[blk * 64 + i * 8 + j] * mb[i * 8 + j];
        }
        acc = __builtin_amdgcn_wmma_f32_16x16x4_f32(
            false, cmat[k], false, bv, (short)0, acc, false, false);
    }

    // Relayout P: C/D layout -> A layout through padded LDS.
    float* pb = &pbuf[wv * 272];
#pragma unroll
    for (int r = 0; r < 8; r++)
        pb[(r + 8 * half) * 17 + ln] = acc[r];
    __syncthreads();

    // Stage 2: D = P x diag(Ca,Ca)
    v8f acc2 = {0.f, 0.f, 0.f, 0.f, 0.f, 0.f, 0.f, 0.f};
#pragma unroll
    for (int k = 0; k < 4; k++) {
        int c0 = 4 * k + 2 * half;
        v2f av;
        av.x = pb[ln * 17 + c0];
        av.y = pb[ln * 17 + c0 + 1];
        acc2 = __builtin_amdgcn_wmma_f32_16x16x4_f32(
            false, av, false, cmat[k], (short)0, acc2, false, false);
    }

    // Epilogue: unpatchify and store (normalized to /255).
    if (valid) {
        int blkPerRow = Wpix >> 3;
        int blk  = n0 + 2 * half + (ln >> 3);
        int brow = blk / blkPerRow, bcol = blk % blkPerRow;
        int j = ln & 7;
        long base = (long)b * batchStride + (long)(brow * 8) * Wpix + bcol * 8 + j;
#pragma unroll
        for (int r = 0; r < 8; r++) {
            float val = (0.25f * acc2[r] + 128.0f) * (1.0f / 255.0f);
            long o = base + (long)r * Wpix;
            outp[o] = val;
            if (replicate) {
                outp[o + chStride] = val;
                outp[o + 2 * chStride] = val;
            }
        }
    }
}

// ---------------------------------------------------------------------------
// Kernel 4: 2x bilinear chroma upsample (half-pixel centers, edge clamp)
// fused with YCbCr->RGB. planB: y already replicated into out channels (RMW).
// ---------------------------------------------------------------------------
__global__ void __launch_bounds__(256)
combine_kernel(const float* __restrict__ ypl,
               const float* __restrict__ cbpl,
               const float* __restrict__ crpl,
               float* __restrict__ out,
               int H, int W, long total, int planB) {
    long idx = (long)blockIdx.x * blockDim.x + threadIdx.x;
    if (idx >= total) return;
    int x = (int)(idx % W);
    long t = idx / W;
    int y = (int)(t % H);
    int b = (int)(t / H);
    int hc = H >> 1, wc = W >> 1;

    int jlo = (y - 1) >> 1;                 // arithmetic shift: y=0 -> -1
    float fy = (y & 1) ? 0.25f : 0.75f;
    int jhi = jlo + 1;
    jlo = jlo < 0 ? 0 : jlo;
    jhi = jhi > hc - 1 ? hc - 1 : jhi;

    int ilo = (x - 1) >> 1;
    float fx = (x & 1) ? 0.25f : 0.75f;
    int ihi = ilo + 1;
    ilo = ilo < 0 ? 0 : ilo;
    ihi = ihi > wc - 1 ? wc - 1 : ihi;

    float w00 = (1.0f - fy) * (1.0f - fx);
    float w01 = (1.0f - fy) * fx;
    float w10 = fy * (1.0f - fx);
    float w11 = fy * fx;

    long cbase = (long)b * hc * wc;
    const float* cbp = cbpl + cbase;
    const float* crp = crpl + cbase;
    long o00 = (long)jlo * wc + ilo, o01 = (long)jlo * wc + ihi;
    long o10 = (long)jhi * wc + ilo, o11 = (long)jhi * wc + ihi;
    float cb = w00 * cbp[o00] + w01 * cbp[o01] + w10 * cbp[o10] + w11 * cbp[o11];
    float cr = w00 * crp[o00] + w01 * crp[o01] + w10 * crp[o10] + w11 * crp[o11];
    float cbs = cb - 0.5f;
    float crs = cr - 0.5f;

    long cs = (long)H * W;
    long o0 = ((long)b * 3 * H + y) * (long)W + x;
    float yv = planB ? out[o0] : ypl[((long)b * H + y) * (long)W + x];

    out[o0]          = yv + 1.403f * crs;
    out[o0 + cs]     = yv - 0.344f * cbs - 0.714f * crs;
    out[o0 + 2 * cs] = yv + 1.773f * cbs;
}

// ---------------------------------------------------------------------------
extern "C" void kernel_launch(void* const* d_in, const int* in_sizes, int n_in,
                              void* d_out, int out_size, void* d_ws, size_t ws_size,
                              hipStream_t stream) {
    (void)n_in; (void)out_size;
    const float* in_y  = (const float*)d_in[0];
    const float* in_cb = (const float*)d_in[1];
    const float* in_cr = (const float*)d_in[2];
    const float* q     = (const float*)d_in[3];
    const float* qty   = (const float*)d_in[4];
    const float* qtc   = (const float*)d_in[5];

    int B = in_sizes[3];
    long hw = (long)in_sizes[0] / B;           // H*W (Ny*64)
    int H = (int)(sqrt((double)hw) + 0.5);     // square images (harness: 512x512)
    int W = H;
    int Ny = (int)(hw / 64);
    int Nc = in_sizes[1] / (B * 64);
    int hc = H / 2, wc = W / 2;

    float* ws    = (float*)d_ws;
    float* multY = ws;
    float* multC = ws + (size_t)B * 64;
    size_t off   = (size_t)B * 64 * 2;
    size_t ypN   = (size_t)B * H * W;
    size_t cpN   = (size_t)B * hc * wc;
    size_t needA = (off + ypN + 2 * cpN) * sizeof(float);

    int planB = (ws_size < needA) ? 1 : 0;
    float *ypl = nullptr, *cbpl, *crpl;
    if (!planB) { ypl = ws + off; cbpl = ypl + ypN; crpl = cbpl + cpN; }
    else        { cbpl = ws + off; crpl = cbpl + cpN; }
    float* out = (float*)d_out;

    // 0) dequant multiplier tables
    mult_kernel<<<B, 128, 0, stream>>>(q, qty, qtc, multY, multC);

    // 1) Y plane IDCT (WMMA)
    long tilesY = (long)B * (Ny / 4);
    int grY = (int)((tilesY + 7) / 8);
    if (planB)
        idct8_wmma_kernel<<<grY, 256, 0, stream>>>(in_y, multY, out, Ny, W,
            (long)3 * H * W, 1, (long)H * W, tilesY);
    else
        idct8_wmma_kernel<<<grY, 256, 0, stream>>>(in_y, multY, ypl, Ny, W,
            (long)H * W, 0, 0, tilesY);

    // 2/3) Cb, Cr quarter planes
    long tilesC = (long)B * (Nc / 4);
    int grC = (int)((tilesC + 7) / 8);
    idct8_wmma_kernel<<<grC, 256, 0, stream>>>(in_cb, multC, cbpl, Nc, wc,
        (long)hc * wc, 0, 0, tilesC);
    idct8_wmma_kernel<<<grC, 256, 0, stream>>>(in_cr, multC, crpl, Nc, wc,
        (long)hc * wc, 0, 0, tilesC);

    // 4) upsample + color convert
    long tot = (long)B * H * W;
    int grD = (int)((tot + 255) / 256);
    combine_kernel<<<grD, 256, 0, stream>>>(ypl, cbpl, crpl, out, H, W, tot, planB);
}